// CustomTokensLayer_4518305595509
// MI455X (gfx1250) — compile-verified
//
#include <hip/hip_runtime.h>

#define VOCAB    128000
#define DIM      2048
#define NTRAIN   256
#define BLOCK    256
#define ROWS     4        // rows (tokens) processed per workgroup
#define MAXMATCH 32

// ---- CDNA5 async Global<->LDS path detection -------------------------------
#if defined(__has_builtin)
#if __has_builtin(__builtin_amdgcn_global_load_async_to_lds_b128) && \
    __has_builtin(__builtin_amdgcn_global_store_async_from_lds_b128)
#define HAVE_ASYNC_LDS 1
#endif
#endif

// Builtin parameter types (confirmed via hipcc diagnostics): int4* in AS1/AS3.
typedef int v4i __attribute__((ext_vector_type(4)));
typedef __attribute__((address_space(1))) v4i gv4i;  // global int4
typedef __attribute__((address_space(3))) v4i lv4i;  // LDS int4

#define CPOL_RT 0   // default temporal
#define CPOL_NT 1   // TH_NT: streaming store, don't pollute caches

__device__ __forceinline__ void wait_asynccnt0() {
#if defined(__has_builtin) && __has_builtin(__builtin_amdgcn_s_wait_asynccnt)
  __builtin_amdgcn_s_wait_asynccnt(0);
#else
  asm volatile("s_wait_asynccnt 0" ::: "memory");
#endif
}

// Each workgroup streams ROWS embedding rows (8KB each) HBM -> LDS -> HBM via
// the CDNA5 async DMA path, patching sparse delta rows in LDS.
__global__ __launch_bounds__(BLOCK) void gather_rows_kernel(
    const int*   __restrict__ x,       // [BATCH*SEQ] token ids
    const float* __restrict__ W,       // [VOCAB, DIM]
    const float* __restrict__ delta,   // [DIM*NTRAIN], column-major over (k,d)
    const int*   __restrict__ tokidx,  // [NTRAIN]
    float*       __restrict__ out)     // [BATCH*SEQ, DIM]
{
  __shared__ float rows[ROWS][DIM];          // 32 KB staging
  __shared__ int   nmatch[ROWS];
  __shared__ int   matchk[ROWS][MAXMATCH];

  const int       t    = threadIdx.x;            // 0..255
  const long long base = (long long)blockIdx.x * ROWS;

  int toks[ROWS];
#pragma unroll
  for (int r = 0; r < ROWS; ++r) toks[r] = x[base + r];  // uniform scalar loads

  if (t < ROWS) nmatch[t] = 0;

  // ---- Issue all async DMA loads up front: 8 outstanding b128 per thread ---
#pragma unroll
  for (int r = 0; r < ROWS; ++r) {
    const float* src = W + (long long)toks[r] * DIM;
#if HAVE_ASYNC_LDS
    __builtin_amdgcn_global_load_async_to_lds_b128(
        (gv4i*)(src + t * 4),        (lv4i*)(&rows[r][t * 4]),        0, CPOL_RT);
    __builtin_amdgcn_global_load_async_to_lds_b128(
        (gv4i*)(src + 1024 + t * 4), (lv4i*)(&rows[r][1024 + t * 4]), 0, CPOL_RT);
#else
    *(float4*)(&rows[r][t * 4])        = *(const float4*)(src + t * 4);
    *(float4*)(&rows[r][1024 + t * 4]) = *(const float4*)(src + 1024 + t * 4);
#endif
  }

  __syncthreads();  // publish nmatch init (and fallback row writes)

  // ---- Sparse-delta match detection (overlaps with in-flight DMA) ----------
  // NTRAIN == BLOCK: each thread owns one trained-token slot; duplicated
  // token_indices entries each contribute once (matches JAX .at[].add()).
  {
    const int myidx = tokidx[t];
#pragma unroll
    for (int r = 0; r < ROWS; ++r) {
      if (myidx == toks[r]) {
        int p = atomicAdd(&nmatch[r], 1);
        if (p < MAXMATCH) matchk[r][p] = t;
      }
    }
  }

#if HAVE_ASYNC_LDS
  wait_asynccnt0();   // per-wave: own async loads landed in LDS
#endif
  __syncthreads();    // cross-wave: all rows resident + match lists final

  // ---- Apply delta rows (rare: expected ~0.2% of rows) ---------------------
#pragma unroll
  for (int r = 0; r < ROWS; ++r) {
    int nm = nmatch[r];
    if (nm > MAXMATCH) nm = MAXMATCH;
    for (int m = 0; m < nm; ++m) {
      const int k = matchk[r][m];
#pragma unroll
      for (int c = 0; c < DIM / BLOCK; ++c) {
        const int d = c * BLOCK + t;
        rows[r][d] += delta[(long long)d * NTRAIN + k];  // delta_rows[k, d]
      }
    }
  }
  __syncthreads();    // DS writes complete before async stores read LDS

  // ---- Async DMA stores, non-temporal (streaming output) -------------------
#pragma unroll
  for (int r = 0; r < ROWS; ++r) {
    float* dst = out + (base + r) * DIM;
#if HAVE_ASYNC_LDS
    __builtin_amdgcn_global_store_async_from_lds_b128(
        (gv4i*)(dst + t * 4),        (lv4i*)(&rows[r][t * 4]),        0, CPOL_NT);
    __builtin_amdgcn_global_store_async_from_lds_b128(
        (gv4i*)(dst + 1024 + t * 4), (lv4i*)(&rows[r][1024 + t * 4]), 0, CPOL_NT);
#else
    *(float4*)(dst + t * 4)        = *(const float4*)(&rows[r][t * 4]);
    *(float4*)(dst + 1024 + t * 4) = *(const float4*)(&rows[r][1024 + t * 4]);
#endif
  }
  // s_endpgm performs the implicit wait-idle for outstanding ASYNCcnt stores.
}

extern "C" void kernel_launch(void* const* d_in, const int* in_sizes, int n_in,
                              void* d_out, int out_size, void* d_ws, size_t ws_size,
                              hipStream_t stream) {
  (void)n_in; (void)out_size; (void)d_ws; (void)ws_size;
  const int*   x      = (const int*)d_in[0];    // [BATCH*SEQ]
  const float* W      = (const float*)d_in[1];  // [VOCAB*DIM]
  const float* delta  = (const float*)d_in[2];  // [NTRAIN*DIM] (col-major)
  const int*   tokidx = (const int*)d_in[3];    // [NTRAIN]
  float*       out    = (float*)d_out;

  const int nrows   = in_sizes[0];          // BATCH*SEQ = 16384
  const int nblocks = (nrows + ROWS - 1) / ROWS;  // 4096
  gather_rows_kernel<<<dim3(nblocks), dim3(BLOCK), 0, stream>>>(
      x, W, delta, tokidx, out);
}